// GloVe_46325517255235
// MI455X (gfx1250) — compile-verified
//
#include <hip/hip_runtime.h>
#include <hip/hip_bf16.h>

typedef __attribute__((ext_vector_type(2))) float v2f;
typedef __attribute__((ext_vector_type(8))) float v8f;

#define TILE 128
#define LDSS 132   // padded LDS row stride (floats): 132 % 64 = 4 -> conflict-free b64 frag reads

__global__ void glove_zero_kernel(float* out) {
    if (threadIdx.x == 0) out[0] = 0.0f;
}

// Fused GloVe loss: sum(Xw * (w@wc^T + b[v] + bc[v] - log X)^2)
// One block computes a 128x128 tile of the implicit V x V hypothesis matrix.
__global__ __launch_bounds__(256) void glove_fused_kernel(
    const float* __restrict__ Xw, const float* __restrict__ X,
    const float* __restrict__ w,  const float* __restrict__ wc,
    const float* __restrict__ b,  const float* __restrict__ bc,
    float* __restrict__ out, int V, int D)
{
    __shared__ float sW[TILE * LDSS];
    __shared__ float sC[TILE * LDSS];
    __shared__ float sred[8];

    const int tid  = threadIdx.x;      // 0..255, 8 waves
    const int lane = tid & 31;
    const int wave = tid >> 5;         // 0..7
    const int half = lane >> 4;        // 0 or 1 (K-pair selector for f32 WMMA frags)
    const int l16  = lane & 15;
    const int m0   = blockIdx.y * TILE;
    const int n0   = blockIdx.x * TILE;

    // ---- cooperative stage: 128 rows x D floats of w and wc, float4-coalesced ----
    const int chunksPerRow = D >> 2;                 // 32 for D=128
    const int total = TILE * chunksPerRow;           // 4096 float4 chunks
    for (int idx = tid; idx < total; idx += 256) {
        int row = idx / chunksPerRow;
        int c4  = (idx % chunksPerRow) << 2;
        float4 vw = *reinterpret_cast<const float4*>(&w [(size_t)(m0 + row) * D + c4]);
        float4 vc = *reinterpret_cast<const float4*>(&wc[(size_t)(n0 + row) * D + c4]);
        *reinterpret_cast<float4*>(&sW[row * LDSS + c4]) = vw;
        *reinterpret_cast<float4*>(&sC[row * LDSS + c4]) = vc;
    }
    __syncthreads();

    // ---- WMMA: each wave owns 16 rows (wave*16..) x 128 cols -> 8 16x16 tiles ----
    // f32 16x16x4 A layout: lane L<16 -> M=L, K={0,1}; lane>=16 -> M=L-16, K={2,3}
    // B (4x16) layout mirrors it: N=L%16, K pair by lane half. Same LDS addressing.
    const int arow = wave * 16 + l16;
    v8f acc[8] = {};                                  // 64 VGPRs of accumulators

    const int ksteps = D >> 2;                        // 32
    #pragma unroll 4
    for (int kk = 0; kk < ksteps; ++kk) {
        const int kbase = (kk << 2) + (half << 1);
        v2f afrag = *reinterpret_cast<const v2f*>(&sW[arow * LDSS + kbase]);
        #pragma unroll
        for (int n = 0; n < 8; ++n) {
            v2f bfrag = *reinterpret_cast<const v2f*>(&sC[(n * 16 + l16) * LDSS + kbase]);
            acc[n] = __builtin_amdgcn_wmma_f32_16x16x4_f32(
                false, afrag, false, bfrag, (short)0, acc[n], false, false);
        }
    }

    // ---- epilogue: C layout = VGPR r, lane -> (M = r + 8*half, N = l16) ----
    float bias[8];
    #pragma unroll
    for (int r = 0; r < 8; ++r) {
        int gm = m0 + wave * 16 + half * 8 + r;
        bias[r] = b[gm] + bc[gm];
    }

    float partial = 0.0f;
    #pragma unroll
    for (int n = 0; n < 8; ++n) {
        const int gn = n0 + n * 16 + l16;
        #pragma unroll
        for (int r = 0; r < 8; ++r) {
            const int gm = m0 + wave * 16 + half * 8 + r;
            const size_t off = (size_t)gm * (size_t)V + (size_t)gn;
            float xw = Xw[off];
            float x  = X[off];
            float diff = acc[n][r] + bias[r] - __logf(x);
            partial = fmaf(xw * diff, diff, partial);
        }
    }

    // ---- reduction: wave32 shuffle -> LDS -> one atomic per block ----
    #pragma unroll
    for (int off = 16; off > 0; off >>= 1)
        partial += __shfl_down(partial, off, 32);
    if (lane == 0) sred[wave] = partial;
    __syncthreads();
    if (tid == 0) {
        float s = 0.0f;
        #pragma unroll
        for (int i = 0; i < 8; ++i) s += sred[i];
        atomicAdd(out, s);
    }
}

extern "C" void kernel_launch(void* const* d_in, const int* in_sizes, int n_in,
                              void* d_out, int out_size, void* d_ws, size_t ws_size,
                              hipStream_t stream) {
    const float* Xw = (const float*)d_in[0];
    const float* X  = (const float*)d_in[1];
    const float* w  = (const float*)d_in[2];
    const float* wc = (const float*)d_in[3];
    const float* b  = (const float*)d_in[4];
    const float* bc = (const float*)d_in[5];
    float* out = (float*)d_out;

    const int V = in_sizes[4];          // b is [V,1]
    const int D = in_sizes[2] / V;      // w is [V,D] -> 128

    glove_zero_kernel<<<1, 32, 0, stream>>>(out);
    dim3 grid(V / TILE, V / TILE);
    glove_fused_kernel<<<grid, 256, 0, stream>>>(Xw, X, w, wc, b, bc, out, V, D);
}